// SpatialAttention_5566277616516
// MI455X (gfx1250) — compile-verified
//
#include <hip/hip_runtime.h>
#include <math.h>

// Shapes from the reference
#define B_N   256   // sites (attention axis)
#define T_N   16
#define L_N   4
#define D_N   64
#define H_N   8
#define KD_N  32
#define VD_N  32
#define TL_N  (T_N * L_N)     // 64
#define HE_N  (H_N * KD_N)    // 256  (fused head*keydim contraction axis)
#define MTOT  (B_N * TL_N)    // 16384 flattened (b,t,l) rows

typedef __attribute__((ext_vector_type(2))) float v2f;
typedef __attribute__((ext_vector_type(8))) float v8f;

#define WMMA_F32(A, Bv, C) \
    __builtin_amdgcn_wmma_f32_16x16x4_f32(false, (A), false, (Bv), (short)0, (C), false, false)

// ---------------------------------------------------------------------------
// Build TRANSPOSED effective weights so GEMM B-operands load as contiguous
// 8-byte pairs:
//   WqT[he][d] = Wq[d][he] * g[h] * scale      (256 x 64)
//   WkT[he][d] = Wk[d][he]                     (256 x 64)
//   WvT[e][d]  = sum_h Wv[d][h][e]             ( 32 x 64)
// ---------------------------------------------------------------------------
__global__ void prep_weights(const float* __restrict__ Wq,
                             const float* __restrict__ Wk,
                             const float* __restrict__ Wv,
                             const float* __restrict__ g,
                             float* __restrict__ WqT,
                             float* __restrict__ WkT,
                             float* __restrict__ WvT) {
    const float scale = 0.17677669529663687f;  // 1/sqrt(KD=32)
    int tid = blockIdx.x * blockDim.x + threadIdx.x;
    int stride = blockDim.x * gridDim.x;
    for (int i = tid; i < HE_N * D_N; i += stride) {
        int he = i / D_N, d = i % D_N;
        int h = he / KD_N;
        WqT[i] = Wq[d * HE_N + he] * g[h] * scale;
        WkT[i] = Wk[d * HE_N + he];
    }
    for (int i = tid; i < VD_N * D_N; i += stride) {
        int e = i / D_N, d = i % D_N;
        float s = 0.0f;
        for (int h = 0; h < H_N; ++h) s += Wv[(d * H_N + h) * VD_N + e];
        WvT[i] = s;
    }
}

// ---------------------------------------------------------------------------
// Y[M x N] = X[M x 64] @ WT[N x 64]^T with V_WMMA_F32_16X16X4_F32.
// One wave per 16x16 tile.  N is a template constant so BOTH the fully
// unrolled K loop and the 8 output stores use immediate offsets only.
// ---------------------------------------------------------------------------
template <int N>
__global__ __launch_bounds__(32) void proj_gemm(const float* __restrict__ X,
                                                const float* __restrict__ WT,
                                                float* __restrict__ Y) {
    const int lane = threadIdx.x;
    const int half = lane >> 4;       // 0: K pair {0,1}, 1: {2,3}
    const int l16  = lane & 15;
    const int m0   = blockIdx.x * 16;
    const int n0   = blockIdx.y * 16;

    const float* xp = X  + (m0 + l16) * D_N + half * 2;   // A row (M = l16)
    const float* wp = WT + (n0 + l16) * D_N + half * 2;   // B col (N = l16)

    v8f acc = {};
#pragma unroll
    for (int k0 = 0; k0 < D_N; k0 += 4) {
        v2f a = *(const v2f*)(xp + k0);
        v2f b = *(const v2f*)(wp + k0);
        acc = WMMA_F32(a, b, acc);
    }

    float* yp = Y + (m0 + half * 8) * N + n0 + l16;       // immediate-offset stores
#pragma unroll
    for (int v = 0; v < 8; ++v)
        yp[v * N] = acc[v];
}

// ---------------------------------------------------------------------------
// Per (t,l) and 16-row tile of query nodes:
//   S = Qg_tile[16x256] @ Kf_{t,l}^T   (16 N-tiles x 64 K-steps of WMMA)
//   p = softmax(S, axis=-1)            -> p_out [T,L,B,B] (non-temporal) + LDS
//   attn = p[16x256] @ Vs_{t,l}[256x32]
// Flat (b,t,l) row = b*64 + tl.
// ---------------------------------------------------------------------------
__global__ __launch_bounds__(32) void attn_kernel(const float* __restrict__ Qg,
                                                  const float* __restrict__ Kf,
                                                  const float* __restrict__ Vs,
                                                  float* __restrict__ attn_out,
                                                  float* __restrict__ p_out) {
    __shared__ float lds_p[16][B_N + 1];  // stride 257 -> conflict-free A reads

    const int lane = threadIdx.x;
    const int half = lane >> 4;
    const int l16  = lane & 15;
    const int m0   = blockIdx.x * 16;     // query-node tile base
    const int tl   = blockIdx.y;          // 0..63 = t*L + l

    // ---- scores: 16 accumulators = S[16 x 256] for this wave ----
    // N-tile stride between Kf row groups: 16 rows * 64 * 256 floats = 1 MiB.
    // nt' in [0,7] -> byte offsets nt'*1MiB + {0,16,32,48} fit the signed
    // 24-bit instruction offset, so TWO base pointers cover all 64 loads
    // per chunk with immediates (3 pointer bumps per 64 WMMAs).
    constexpr int NT_STRIDE = 16 * TL_N * HE_N;  // floats (1 MiB in bytes)

    v8f acc[16] = {};
    const float* qp  = Qg + ((m0 + l16) * TL_N + tl) * HE_N + half * 2;
    const float* kb0 = Kf + ((size_t)l16 * TL_N + tl) * HE_N + half * 2;
    const float* kb1 = kb0 + (size_t)8 * NT_STRIDE;

    for (int kc = 0; kc < HE_N / 16; ++kc) {   // 16 chunks x (4 WMMA x 16 tiles)
        v2f a0 = *(const v2f*)(qp + 0);
        v2f a1 = *(const v2f*)(qp + 4);
        v2f a2 = *(const v2f*)(qp + 8);
        v2f a3 = *(const v2f*)(qp + 12);
        qp += 16;
#pragma unroll
        for (int nt = 0; nt < 16; ++nt) {
            const float* kb = (nt < 8 ? kb0 : kb1) + (size_t)(nt & 7) * NT_STRIDE;
            v2f b0 = *(const v2f*)(kb + 0);
            v2f b1 = *(const v2f*)(kb + 4);
            v2f b2 = *(const v2f*)(kb + 8);
            v2f b3 = *(const v2f*)(kb + 12);
            v8f c = acc[nt];
            c = WMMA_F32(a0, b0, c);
            c = WMMA_F32(a1, b1, c);
            c = WMMA_F32(a2, b2, c);
            c = WMMA_F32(a3, b3, c);
            acc[nt] = c;
        }
        kb0 += 16;
        kb1 += 16;
    }

    // ---- softmax over 256 source nodes (row M = half*8 + v lives in one half) ----
    float rmax[8], rinv[8];
#pragma unroll
    for (int v = 0; v < 8; ++v) {
        float m = acc[0][v];
#pragma unroll
        for (int nt = 1; nt < 16; ++nt) m = fmaxf(m, acc[nt][v]);
        for (int off = 1; off < 16; off <<= 1)
            m = fmaxf(m, __shfl_xor(m, off, 16));
        rmax[v] = m;
    }
#pragma unroll
    for (int v = 0; v < 8; ++v) {
        float s = 0.0f;
#pragma unroll
        for (int nt = 0; nt < 16; ++nt) {
            float e = __expf(acc[nt][v] - rmax[v]);
            acc[nt][v] = e;
            s += e;
        }
        for (int off = 1; off < 16; off <<= 1)
            s += __shfl_xor(s, off, 16);
        rinv[v] = 1.0f / s;
    }

    // ---- normalize; emit p (global non-temporal + LDS) ----
    // p_out flat: ((tl*256 + row)*256 + col); single base, immediate offsets.
    float* pb = p_out + ((size_t)tl * B_N + m0 + half * 8) * B_N + l16;
#pragma unroll
    for (int v = 0; v < 8; ++v) {
        const int M = half * 8 + v;
#pragma unroll
        for (int nt = 0; nt < 16; ++nt) {
            float pv = acc[nt][v] * rinv[v];
            lds_p[M][nt * 16 + l16] = pv;
            __builtin_nontemporal_store(pv, pb + v * B_N + nt * 16);
        }
    }
    __syncthreads();

    // ---- attn = p @ Vs_{t,l} : 16 x 32 output (two 16-wide N tiles) ----
    // Vs row k (source node) lives at (k*64 + tl)*32; row stride 2048 floats
    // (8 KiB).  Max offset 255*8KiB ~= 2.06 MiB < 8 MiB immediate range, so a
    // single fixed base covers the fully unrolled loop with zero address math.
    v8f o0 = {}, o1 = {};
    const float* vb = Vs + ((size_t)(half * 2) * TL_N + tl) * VD_N + l16;
    const float* pr = &lds_p[l16][half * 2];
#pragma unroll
    for (int k0 = 0; k0 < B_N; k0 += 4) {
        v2f a;
        a.x = pr[k0];
        a.y = pr[k0 + 1];
        const float* r0 = vb + (size_t)k0 * (TL_N * VD_N);
        const float* r1 = r0 + (TL_N * VD_N);
        v2f b0, b1;
        b0.x = r0[0];  b0.y = r1[0];
        b1.x = r0[16]; b1.y = r1[16];
        o0 = WMMA_F32(a, b0, o0);
        o1 = WMMA_F32(a, b1, o1);
    }

    // attn_out [B,T,L,VD]: base + immediate offsets (v*64*32 floats)
    float* ob = attn_out + ((size_t)(m0 + half * 8) * TL_N + tl) * VD_N + l16;
#pragma unroll
    for (int v = 0; v < 8; ++v) {
        ob[v * TL_N * VD_N]      = o0[v];
        ob[v * TL_N * VD_N + 16] = o1[v];
    }
}

// ---------------------------------------------------------------------------
extern "C" void kernel_launch(void* const* d_in, const int* in_sizes, int n_in,
                              void* d_out, int out_size, void* d_ws, size_t ws_size,
                              hipStream_t stream) {
    const float* queries = (const float*)d_in[0];  // [B,T,L,D]
    const float* keys    = (const float*)d_in[1];
    const float* values  = (const float*)d_in[2];
    const float* Wq      = (const float*)d_in[3];  // [D,H,KD]
    const float* Wk      = (const float*)d_in[4];
    const float* Wv      = (const float*)d_in[5];  // [D,H,VD]
    const float* g       = (const float*)d_in[6];  // [H]

    // Workspace layout (floats): ~35.8 MB total
    float* ws  = (float*)d_ws;
    float* Qg  = ws;                                  // MTOT x 256
    float* Kf  = Qg  + (size_t)MTOT * HE_N;           // MTOT x 256
    float* Vs  = Kf  + (size_t)MTOT * HE_N;           // MTOT x 32
    float* WqT = Vs  + (size_t)MTOT * VD_N;           // 256 x 64 (transposed)
    float* WkT = WqT + (size_t)HE_N * D_N;            // 256 x 64 (transposed)
    float* WvT = WkT + (size_t)HE_N * D_N;            //  32 x 64 (transposed)

    float* attn_out = (float*)d_out;                                  // B*T*L*VD
    float* p_out    = attn_out + (size_t)B_N * T_N * L_N * VD_N;      // T*L*B*B

    prep_weights<<<32, 256, 0, stream>>>(Wq, Wk, Wv, g, WqT, WkT, WvT);

    dim3 gq(MTOT / 16, HE_N / 16);
    proj_gemm<HE_N><<<gq, 32, 0, stream>>>(queries, WqT, Qg);
    proj_gemm<HE_N><<<gq, 32, 0, stream>>>(keys,    WkT, Kf);
    dim3 gv(MTOT / 16, VD_N / 16);
    proj_gemm<VD_N><<<gv, 32, 0, stream>>>(values,  WvT, Vs);

    dim3 ga(16 /*row tiles*/, TL_N /*t*L+l*/);
    attn_kernel<<<ga, 32, 0, stream>>>(Qg, Kf, Vs, attn_out, p_out);
}